// SENNGC_2405181686013
// MI455X (gfx1250) — compile-verified
//
#include <hip/hip_runtime.h>
#include <math.h>

// Problem constants (match reference)
#define NN    128
#define GG    1024          // B * ORDER
#define HH    16
#define KK    10
#define PO    32
#define KH    160           // K * H

// d_out layout (tuple concatenated flat, in return order)
#define P_OFF 0
#define T_OFF (GG * PO)             // 32768
#define A_OFF (T_OFF + GG * KK)     // 43008

typedef float v2f __attribute__((ext_vector_type(2)));
typedef float v8f __attribute__((ext_vector_type(8)));

__device__ __forceinline__ float wave_max(float v) {
#pragma unroll
  for (int off = 16; off >= 1; off >>= 1)
    v = fmaxf(v, __shfl_xor(v, off, 32));
  return v;
}
__device__ __forceinline__ float wave_sum(float v) {
#pragma unroll
  for (int off = 16; off >= 1; off >>= 1)
    v += __shfl_xor(v, off, 32);
  return v;
}

__global__ __launch_bounds__(128)
void sennc_fused(const float* __restrict__ x,        // (G, N)
                 const float* __restrict__ gumbel_u, // (G, N, N)
                 const float* __restrict__ phi,      // (N, N)
                 const float* __restrict__ Wcat_w,   // (H, 2)
                 const float* __restrict__ Wcat_b,   // (H)
                 const float* __restrict__ a_w,      // (H)
                 const float* __restrict__ Wnode_w,  // (H)
                 const float* __restrict__ Wnode_b,  // (H)
                 const float* __restrict__ proj_w,   // (PO, KH)
                 const float* __restrict__ proj_b,   // (PO)
                 float* __restrict__ out)
{
  // LDS: ~52 KB
  __shared__ float s_alpha[4][16][132];   // per-wave alpha chunk, padded (bank-conflict-free frag reads)
  __shared__ float s_np[NN][HH];          // node_proj
  __shared__ float s_emb[NN][17];         // node_emb (padded)
  __shared__ float s_node[NN];
  __shared__ float s_norm[NN];
  __shared__ float s_W1[HH], s_W2[HH], s_bc[HH], s_aw[HH];

  const int g    = blockIdx.x;
  const int tid  = threadIdx.x;
  const int w    = tid >> 5;      // wave id 0..3
  const int lane = tid & 31;
  const int l    = lane & 15;     // WMMA sub-lane
  const int hv   = lane >> 4;     // 0 = lanes 0-15, 1 = lanes 16-31

  // ---- small params + node into LDS ----
  if (tid < HH) {
    s_W1[tid] = Wcat_w[tid * 2 + 0];
    s_W2[tid] = Wcat_w[tid * 2 + 1];
    s_bc[tid] = Wcat_b[tid];
    s_aw[tid] = a_w[tid];
  }
  s_node[tid] = x[g * NN + tid];
  __syncthreads();

  // node_proj[j][h] = node[j]*Wnode_w[h] + Wnode_b[h]
  {
    const float xv = s_node[tid];
#pragma unroll
    for (int h = 0; h < HH; ++h)
      s_np[tid][h] = xv * Wnode_w[h] + Wnode_b[h];
  }
  __syncthreads();

  const size_t gbase = (size_t)g * (NN * NN);

  // ---- per-wave: two 16-row M tiles ----
  for (int t = w; t < 8; t += 4) {
    // 1) fused rows: gumbel -> A softmax (write A) -> scores -> alpha softmax (to LDS)
    for (int r = 0; r < 16; ++r) {
      const int i = t * 16 + r;
      const float xi = s_node[i];
      float z[4], sc[4];
#pragma unroll
      for (int q = 0; q < 4; ++q) {
        const int j = q * 32 + lane;
        const float u   = gumbel_u[gbase + i * NN + j];
        const float gum = -__logf(-__logf(u + 1e-9f) + 1e-9f);
        z[q] = (phi[i * NN + j] + gum) * 2.0f;   // /TEMP, TEMP=0.5
        const float xj = s_node[j];
        float s = 0.f;
#pragma unroll
        for (int h = 0; h < HH; ++h) {
          float v = fmaf(xi, s_W1[h], fmaf(xj, s_W2[h], s_bc[h]));
          v = (v > 0.f) ? v : 0.2f * v;          // leaky_relu 0.2
          s = fmaf(s_aw[h], v, s);
        }
        sc[q] = s;
      }
      // row softmax -> A
      float m1 = fmaxf(fmaxf(z[0], z[1]), fmaxf(z[2], z[3]));
      m1 = wave_max(m1);
      float e[4], s1 = 0.f;
#pragma unroll
      for (int q = 0; q < 4; ++q) { e[q] = __expf(z[q] - m1); s1 += e[q]; }
      s1 = wave_sum(s1);
      const float inv1 = 1.0f / s1;
      float logit[4];
#pragma unroll
      for (int q = 0; q < 4; ++q) {
        const float Av = e[q] * inv1;
        out[A_OFF + gbase + i * NN + q * 32 + lane] = Av;       // A output
        logit[q] = sc[q] + __logf(Av + 1e-8f);                  // scores + log(A+EPS)
      }
      // row softmax -> alpha (into LDS chunk)
      float m2 = fmaxf(fmaxf(logit[0], logit[1]), fmaxf(logit[2], logit[3]));
      m2 = wave_max(m2);
      float ee[4], s2 = 0.f;
#pragma unroll
      for (int q = 0; q < 4; ++q) { ee[q] = __expf(logit[q] - m2); s2 += ee[q]; }
      s2 = wave_sum(s2);
      const float inv2 = 1.0f / s2;
#pragma unroll
      for (int q = 0; q < 4; ++q)
        s_alpha[w][r][q * 32 + lane] = ee[q] * inv2;
    }

    // 2) emb_tile(16x16) = alpha_chunk(16x128) @ node_proj(128x16) via f32 WMMA
    v8f c = {};
    for (int k0 = 0; k0 < NN; k0 += 4) {
      v2f a, b;
      // A 16x4 f32 layout: lanes0-15 K=k0,k0+1 ; lanes16-31 K=k0+2,k0+3 (M = l)
      a.x = s_alpha[w][l][k0 + (hv ? 2 : 0)];
      a.y = s_alpha[w][l][k0 + (hv ? 3 : 1)];
      // B 4x16 f32 (row striped across lanes): lanes0-15 K=k0,k0+1 ; lanes16-31 K=k0+2,k0+3 (N = l)
      b.x = s_np[k0 + (hv ? 2 : 0)][l];
      b.y = s_np[k0 + (hv ? 3 : 1)][l];
      c = __builtin_amdgcn_wmma_f32_16x16x4_f32(false, a, false, b, (short)0, c, false, false);
    }
    // relu + deposit tile: VGPR r -> (M = r + 8*hv, N = l)
#pragma unroll
    for (int r = 0; r < 8; ++r)
      s_emb[t * 16 + r + (hv ? 8 : 0)][l] = fmaxf(c[r], 0.f);
  }
  __syncthreads();

  // ---- norms ----
  {
    float s = 0.f;
#pragma unroll
    for (int h = 0; h < HH; ++h) { const float v = s_emb[tid][h]; s = fmaf(v, v, s); }
    s_norm[tid] = sqrtf(s);
  }
  __syncthreads();

  // ---- top-k (wave 0) + projection ----
  if (w == 0) {
    float vv[4];
#pragma unroll
    for (int q = 0; q < 4; ++q) vv[q] = s_norm[q * 32 + lane];

    int tk[KK];
    int myidx = 0;
#pragma unroll
    for (int k = 0; k < KK; ++k) {
      float bv = -INFINITY; int bi = NN;
#pragma unroll
      for (int q = 0; q < 4; ++q)
        if (vv[q] > bv) { bv = vv[q]; bi = q * 32 + lane; }   // strict > : lowest j wins ties
#pragma unroll
      for (int off = 16; off >= 1; off >>= 1) {
        const float ov = __shfl_xor(bv, off, 32);
        const int   oi = __shfl_xor(bi, off, 32);
        if (ov > bv || (ov == bv && oi < bi)) { bv = ov; bi = oi; }
      }
      tk[k] = bi;                       // uniform across wave after butterfly
      if (lane == k) myidx = bi;
      if ((bi & 31) == lane) vv[bi >> 5] = -INFINITY;  // remove winner
    }
    if (lane < KK) out[T_OFF + g * KK + lane] = (float)myidx;

    // projected[p] = proj_b[p] + sum_{k,h} emb[tk[k]][h] * proj_w[p, k*H+h]
    float acc = proj_b[lane];
#pragma unroll
    for (int k = 0; k < KK; ++k) {
      const int js = tk[k];
#pragma unroll
      for (int h = 0; h < HH; ++h)
        acc = fmaf(s_emb[js][h], proj_w[lane * KH + k * HH + h], acc);
    }
    out[P_OFF + g * PO + lane] = acc;
  }
}

extern "C" void kernel_launch(void* const* d_in, const int* in_sizes, int n_in,
                              void* d_out, int out_size, void* d_ws, size_t ws_size,
                              hipStream_t stream) {
  (void)in_sizes; (void)n_in; (void)out_size; (void)d_ws; (void)ws_size;
  const float* x       = (const float*)d_in[0];
  const float* gu      = (const float*)d_in[1];
  const float* phi     = (const float*)d_in[2];
  const float* Wcat_w  = (const float*)d_in[3];
  const float* Wcat_b  = (const float*)d_in[4];
  const float* a_w     = (const float*)d_in[5];
  const float* Wnode_w = (const float*)d_in[6];
  const float* Wnode_b = (const float*)d_in[7];
  const float* proj_w  = (const float*)d_in[8];
  const float* proj_b  = (const float*)d_in[9];
  float* out = (float*)d_out;

  hipLaunchKernelGGL(sennc_fused, dim3(GG), dim3(128), 0, stream,
                     x, gu, phi, Wcat_w, Wcat_b, a_w, Wnode_w, Wnode_b,
                     proj_w, proj_b, out);
}